// ArcTanDistortion_54365696033609
// MI455X (gfx1250) — compile-verified
//
#include <hip/hip_runtime.h>
#include <cmath>

// y = (2/pi) * atan(GAIN * x) / log(GAIN), elementwise over 67,108,864 fp32.
//
// MI455X roofline: 512 MiB streamed @ 23.3 TB/s  => ~23 us floor. Pure
// bandwidth-bound streaming kernel: 128-bit non-temporal loads/stores
// (CDNA5 TH=NT cache hints; data is touched exactly once so we keep it out
// of the 192 MB L2), wave32-friendly 256-thread blocks, atanf via OCML
// (v_rcp_f32 range reduction + odd polynomial) which sits at/under the
// memory time. WMMA/TDM intentionally unused: no contraction, no reuse.

typedef float v4f __attribute__((ext_vector_type(4)));

__global__ __launch_bounds__(256) void ArcTanDistortion_v4_kernel(
    const v4f* __restrict__ xv, v4f* __restrict__ yv,
    int n4, float gain, float scale) {
  int i = blockIdx.x * blockDim.x + threadIdx.x;
  if (i >= n4) return;

  // global_load_b128 with non-temporal hint (single-use stream)
  v4f v = __builtin_nontemporal_load(xv + i);

  v4f r;
  r.x = atanf(gain * v.x) * scale;
  r.y = atanf(gain * v.y) * scale;
  r.z = atanf(gain * v.z) * scale;
  r.w = atanf(gain * v.w) * scale;

  // global_store_b128 with non-temporal hint (output never re-read)
  __builtin_nontemporal_store(r, yv + i);
}

__global__ __launch_bounds__(64) void ArcTanDistortion_tail_kernel(
    const float* __restrict__ x, float* __restrict__ y,
    int start, int n, float gain, float scale) {
  int i = start + blockIdx.x * blockDim.x + threadIdx.x;
  if (i < n) {
    y[i] = atanf(gain * x[i]) * scale;
  }
}

extern "C" void kernel_launch(void* const* d_in, const int* in_sizes, int n_in,
                              void* d_out, int out_size, void* d_ws, size_t ws_size,
                              hipStream_t stream) {
  (void)n_in; (void)d_ws; (void)ws_size; (void)out_size;

  const float* x = (const float*)d_in[0];
  float* y = (float*)d_out;
  const int n = in_sizes[0];

  const float gain = 67.0f;
  // (2/pi) / ln(67), computed in double on the host, one fp32 constant on device.
  const float scale = (float)((2.0 / M_PI) / std::log(67.0));

  const int n4 = n >> 2;           // number of float4 packets
  const int tail_start = n4 << 2;  // first scalar-tail element

  if (n4 > 0) {
    const int threads = 256;
    const int blocks = (n4 + threads - 1) / threads;  // 65536 for 2^26 elems
    ArcTanDistortion_v4_kernel<<<blocks, threads, 0, stream>>>(
        (const v4f*)x, (v4f*)y, n4, gain, scale);
  }

  const int tail = n - tail_start;
  if (tail > 0) {
    ArcTanDistortion_tail_kernel<<<1, 64, 0, stream>>>(
        x, y, tail_start, n, gain, scale);
  }
}